// GaussianRenderer_7009386627560
// MI455X (gfx1250) — compile-verified
//
#include <hip/hip_runtime.h>
#include <hip/hip_bf16.h>

// Gaussian point-splat renderer for MI455X (gfx1250), compile-only target.
//
// Strategy: counting-sort by pixel (no global sort), then per-pixel
// back-to-front composite with a canonical (z desc, idx asc) order that
// reproduces jnp.lexsort((-z, pix)) stability and makes the atomic
// bucketing deterministic in its final output.
//
// CDNA5-specific paths used:
//   * global_load_async_to_lds_b32 + s_wait_asynccnt (async LDS staging of xyz)
//   * global_prefetch_b8 via __builtin_prefetch in the gather/composite pass
//   * wave32-sized workgroups, LDS block scan

#define RB   8          // batch
#define RN   262144     // gaussians per batch (2^18)
#define RH   512
#define RW   512
#define RHW  (RH * RW)  // 262144
#define TPB  256
#define MAXL 24         // per-pixel in-register sort capacity

// ---------------------------------------------------------------------------
// projection: matches reference (trunc-toward-zero after clip to +-1e9)
// ---------------------------------------------------------------------------
__device__ __forceinline__ int project_pix(float x, float y, float z) {
    float zf = z + 1e-6f;
    float xf = x / zf * 512.0f + 256.0f;   // focal=(H+W)/2=512, +W/2
    float yf = y / zf * 512.0f + 256.0f;   // +H/2
    xf = fminf(fmaxf(xf, -1e9f), 1e9f);
    yf = fminf(fmaxf(yf, -1e9f), 1e9f);
    int xi = (int)xf;                      // C cast truncates toward zero
    int yi = (int)yf;
    if (xi < 0 || xi >= RW || yi < 0 || yi >= RH) return -1;
    return yi * RW + xi;
}

// ---------------------------------------------------------------------------
// Async-stage 256 gaussians (768 floats) of xyz into LDS.
// Coalesced: async instruction k moves floats [k*256 .. k*256+255].
// LDS offset = low 32 bits of the flat address (LDS aperture keeps the
// byte offset in addr[31:0] per CDNA5 ISA 10.2).
// Each wave waits its own ASYNCcnt, then the workgroup barrier publishes LDS.
// ---------------------------------------------------------------------------
__device__ __forceinline__ void stage_xyz_tile(const float* __restrict__ src,
                                               float* tile, int t) {
#pragma unroll
    for (int k = 0; k < 3; ++k) {
        unsigned lds_addr = (unsigned)(size_t)&tile[k * TPB + t];
        const float* gptr = src + k * TPB + t;
        asm volatile("global_load_async_to_lds_b32 %0, %1, off"
                     :: "v"(lds_addr), "v"(gptr)
                     : "memory");
    }
    asm volatile("s_wait_asynccnt 0" ::: "memory");
    __syncthreads();
}

// ---------------------------------------------------------------------------
// Pass 1: per-pixel histogram
// ---------------------------------------------------------------------------
__global__ void gr_count_kernel(const float* __restrict__ xyz,
                                unsigned* __restrict__ counts) {
    __shared__ float tile[3 * TPB];
    int t = threadIdx.x;
    size_t gbase = (size_t)blockIdx.x * TPB;      // global gaussian id base
    stage_xyz_tile(xyz + gbase * 3, tile, t);
    float x = tile[3 * t + 0];
    float y = tile[3 * t + 1];
    float z = tile[3 * t + 2];
    int pix = project_pix(x, y, z);
    if (pix >= 0) {
        int b = (int)(gbase >> 18);               // RN = 2^18, block within batch
        atomicAdd(&counts[(size_t)b * RHW + pix], 1u);
    }
}

// ---------------------------------------------------------------------------
// Pass 2: exclusive scan of 262144 counts per batch.
// One 1024-thread workgroup per batch; 256 segments of 1024 with LDS
// Hillis-Steele scan and a running carry.
// ---------------------------------------------------------------------------
__global__ void gr_scan_kernel(const unsigned* __restrict__ counts,
                               unsigned* __restrict__ bstart) {
    __shared__ unsigned buf[2][1024];
    int b = blockIdx.x;
    int t = threadIdx.x;
    const unsigned* cb = counts + (size_t)b * RHW;
    unsigned* ob = bstart + (size_t)b * RHW;
    unsigned carry = 0;
    for (int seg = 0; seg < RHW / 1024; ++seg) {
        unsigned v = cb[seg * 1024 + t];
        int src = 0;
        buf[0][t] = v;
        __syncthreads();
#pragma unroll
        for (int off = 1; off < 1024; off <<= 1) {
            unsigned x = buf[src][t];
            if (t >= off) x += buf[src][t - off];
            buf[src ^ 1][t] = x;
            src ^= 1;
            __syncthreads();
        }
        unsigned incl  = buf[src][t];
        unsigned total = buf[src][1023];
        ob[seg * 1024 + t] = carry + incl - v;    // exclusive prefix
        carry += total;
        __syncthreads();                          // protect buf before next seg
    }
}

// ---------------------------------------------------------------------------
// Pass 3: scatter gaussian indices into per-pixel buckets.
// bstart is consumed as a cursor: after this pass bstart[p] == bucket end,
// so composite recovers start = end - count (no third array needed).
// ---------------------------------------------------------------------------
__global__ void gr_scatter_kernel(const float* __restrict__ xyz,
                                  unsigned* __restrict__ bstart,
                                  unsigned* __restrict__ indices) {
    __shared__ float tile[3 * TPB];
    int t = threadIdx.x;
    size_t gbase = (size_t)blockIdx.x * TPB;
    stage_xyz_tile(xyz + gbase * 3, tile, t);
    float x = tile[3 * t + 0];
    float y = tile[3 * t + 1];
    float z = tile[3 * t + 2];
    int pix = project_pix(x, y, z);
    if (pix >= 0) {
        size_t gid = gbase + t;
        int b = (int)(gid >> 18);
        unsigned i = (unsigned)(gid & (RN - 1));
        unsigned pos = atomicAdd(&bstart[(size_t)b * RHW + pix], 1u);
        indices[(size_t)b * RN + pos] = i;
    }
}

// ---------------------------------------------------------------------------
// Pass 4: per-pixel sort + back-to-front alpha composite.
// Order: z descending, tie -> smaller original index first (stable lexsort).
// a_k = o_k * T_prev,  T *= max(1-o, 1e-12)  (reference's log clip).
// depth = z of the last composited splat (front-most).
// ---------------------------------------------------------------------------
__global__ void gr_composite_kernel(const float* __restrict__ xyz,
                                    const float* __restrict__ opacity,
                                    const float* __restrict__ color,
                                    const unsigned* __restrict__ counts,
                                    const unsigned* __restrict__ bend,
                                    const unsigned* __restrict__ indices,
                                    float* __restrict__ out) {
    int p = blockIdx.x * blockDim.x + threadIdx.x;  // 0 .. RB*RHW-1
    int b   = p >> 18;
    int pix = p & (RHW - 1);
    unsigned cnt   = counts[p];
    unsigned end   = bend[p];
    unsigned start = end - cnt;
    const size_t bn = (size_t)b * RN;

    float T = 1.0f, cr = 0.0f, cg = 0.0f, cb = 0.0f, depth = 0.0f;

    if (cnt <= MAXL) {
        float zl[MAXL];
        int   il[MAXL];
        for (unsigned j = 0; j < cnt; ++j) {
            int gi = (int)indices[bn + start + j];
            float zj = xyz[(bn + gi) * 3 + 2];
            __builtin_prefetch(color + (bn + gi) * 3, 0, 0);   // global_prefetch_b8
            int k = (int)j;
            while (k > 0 && (zl[k - 1] < zj ||
                             (zl[k - 1] == zj && il[k - 1] > gi))) {
                zl[k] = zl[k - 1];
                il[k] = il[k - 1];
                --k;
            }
            zl[k] = zj;
            il[k] = gi;
        }
        for (unsigned k = 0; k < cnt; ++k) {
            int gi = il[k];
            float o = opacity[bn + gi];
            const float* c = color + (bn + gi) * 3;
            float a = o * T;
            cr += a * c[0];
            cg += a * c[1];
            cb += a * c[2];
            T *= fmaxf(1.0f - o, 1e-12f);
            depth = zl[k];
        }
    } else {
        // Pathological bucket: O(cnt^2) selection in canonical order.
        float pz = __builtin_inff();
        int pidx = -1;
        for (unsigned k = 0; k < cnt; ++k) {
            float bz = -__builtin_inff();
            int bidx = 0x7fffffff;
            bool found = false;
            for (unsigned j = 0; j < cnt; ++j) {
                int gi = (int)indices[bn + start + j];
                float zj = xyz[(bn + gi) * 3 + 2];
                bool after  = (zj < pz) || (zj == pz && gi > pidx);
                bool better = (zj > bz) || (zj == bz && gi < bidx);
                if (after && better) { bz = zj; bidx = gi; found = true; }
            }
            if (!found) break;
            float o = opacity[bn + bidx];
            const float* c = color + (bn + bidx) * 3;
            float a = o * T;
            cr += a * c[0];
            cg += a * c[1];
            cb += a * c[2];
            T *= fmaxf(1.0f - o, 1e-12f);
            depth = bz;
            pz = bz;
            pidx = bidx;
        }
    }

    // render: (B,3,H,W); depth: (B,1,H,W) appended after render.
    out[((size_t)b * 3 + 0) * RHW + pix] = cr;
    out[((size_t)b * 3 + 1) * RHW + pix] = cg;
    out[((size_t)b * 3 + 2) * RHW + pix] = cb;
    out[(size_t)RB * 3 * RHW + (size_t)b * RHW + pix] = depth;
}

// ---------------------------------------------------------------------------
// Launch. Inputs: 0=xyz, 1=scale(unused), 2=rotation(unused), 3=opacity,
// 4=color. Workspace layout (24 MB): counts | bucket cursors | index buckets.
// ---------------------------------------------------------------------------
extern "C" void kernel_launch(void* const* d_in, const int* in_sizes, int n_in,
                              void* d_out, int out_size, void* d_ws, size_t ws_size,
                              hipStream_t stream) {
    const float* xyz     = (const float*)d_in[0];
    const float* opacity = (const float*)d_in[3];
    const float* color   = (const float*)d_in[4];
    float* out = (float*)d_out;

    unsigned* counts  = (unsigned*)d_ws;                  //  8 MB
    unsigned* bstart  = counts + (size_t)RB * RHW;        //  8 MB
    unsigned* indices = bstart + (size_t)RB * RHW;        //  8 MB
    (void)in_sizes; (void)n_in; (void)out_size; (void)ws_size;

    hipMemsetAsync(counts, 0, (size_t)RB * RHW * sizeof(unsigned), stream);

    int nblk = (RB * RN) / TPB;                           // 8192 blocks
    gr_count_kernel<<<nblk, TPB, 0, stream>>>(xyz, counts);
    gr_scan_kernel<<<RB, 1024, 0, stream>>>(counts, bstart);
    gr_scatter_kernel<<<nblk, TPB, 0, stream>>>(xyz, bstart, indices);
    gr_composite_kernel<<<(RB * RHW) / TPB, TPB, 0, stream>>>(
        xyz, opacity, color, counts, bstart, indices, out);
}